// ChannelAttention_45226005627368
// MI455X (gfx1250) — compile-verified
//
#include <hip/hip_runtime.h>

#define B_   32
#define N_   784
#define C_   768
#define G_   8
#define CH_  96
#define C3_  2304

typedef __attribute__((ext_vector_type(2))) float v2f;
typedef __attribute__((ext_vector_type(8))) float v8f;
typedef __attribute__((ext_vector_type(4))) unsigned int u32x4;
typedef __attribute__((ext_vector_type(4))) int i32x4;
typedef __attribute__((ext_vector_type(8))) int i32x8;

// ---- Tensor Data Mover availability / arity selection ----------------------
#if __has_builtin(__builtin_amdgcn_tensor_load_to_lds) && \
    __has_builtin(__builtin_amdgcn_s_wait_tensorcnt)
#  define USE_TDM 1
#  if __has_include(<hip/amd_detail/amd_gfx1250_TDM.h>) || (__clang_major__ >= 23)
#    define TDM_ARGS6 1
#  endif
#endif

// Compile-time probes (visible in hipcc stderr) so we know which path built.
#ifdef USE_TDM
#  ifdef TDM_ARGS6
#    pragma message("CDNA5 probe: TDM path ENABLED (6-arg tensor_load_to_lds)")
#  else
#    pragma message("CDNA5 probe: TDM path ENABLED (5-arg tensor_load_to_lds)")
#  endif
#else
#  pragma message("CDNA5 probe: TDM builtins NOT available; manual LDS staging")
#endif

#ifdef USE_TDM
// low 32 bits of a generic pointer to LDS == LDS byte offset (flat aperture)
__device__ __forceinline__ unsigned lds_off(const void* p) {
  return (unsigned)(unsigned long long)(uintptr_t)p;
}

// D# group 1 for a [rows x 768] f32 tensor, tile tile_rows x 16 cols, LDS rows
// padded: 16 DWORDs of data then 4 DWORDs of pad (matches float[rows][20]).
__device__ __forceinline__ i32x8 tdm_g1(int tensor_rows, int tile_rows) {
  i32x8 g = {};
  g[0] = (int)((2u << 16)     // data_size = 4B
             | (1u << 20)     // pad_enable
             | (3u << 22)     // pad_interval: 16 DWORDs
             | (3u << 25));   // pad_amount:   4 DWORDs
  g[1] = (int)(768u << 16);                       // tensor_dim0 = 768 (lo16)
  g[2] = (int)((unsigned)tensor_rows << 16);      // tensor_dim1 (lo16)
  g[3] = (int)(16u << 16);                        // tile_dim0 = 16
  g[4] = tile_rows;                               // tile_dim1
  g[5] = 768;                                     // tensor_dim0_stride
  return g;
}

__device__ __forceinline__ void tdm_issue(unsigned lds, unsigned long long ga,
                                          i32x8 g1) {
  u32x4 g0;
  g0[0] = 1u;                                     // count = 1 valid descriptor
  g0[1] = lds;                                    // lds_addr
  g0[2] = (unsigned)ga;                           // global_addr[31:0]
  g0[3] = (unsigned)(ga >> 32) | (2u << 30);      // global_addr[56:32]|type=2
  i32x4 z = {};
#ifdef TDM_ARGS6
  __builtin_amdgcn_tensor_load_to_lds(g0, g1, z, z, (i32x8){}, 0);
#else
  __builtin_amdgcn_tensor_load_to_lds(g0, g1, z, z, 0);
#endif
}
#endif  // USE_TDM

// ---------------------------------------------------------------------------
// Generic GEMM: C[M,NC] = A[M,768] @ W[NC,768]^T + bias   (all row-major f32)
// Block tile 64(M) x 128(N), 256 threads = 8 waves in a 2x4 grid; each wave
// computes 32x32 via 4 accumulators -> per k-step 4 ds_load_b64 fragments
// feed 4 V_WMMA_F32_16X16X4_F32 (1:1). K staged through LDS in chunks of 16,
// double-buffered via TDM: wave 0 issues tensor_load_to_lds for chunk k+1
// while all waves compute chunk k (TENSORcnt ordered, barrier publishes LDS).
// ---------------------------------------------------------------------------
template <int NC>
__global__ __launch_bounds__(256) void gemm_bias_kernel(
    const float* __restrict__ A, const float* __restrict__ W,
    const float* __restrict__ bias, float* __restrict__ Cout) {
  constexpr int K = C_;
  __shared__ float As[2][64][20];   // 16 data + 4 pad DWORDs per row
  __shared__ float Ws[2][128][20];

  const int bm = blockIdx.y, bn = blockIdx.x;
  const int tid  = threadIdx.x;
  const int wave = tid >> 5, lane = tid & 31;
  const int half = lane >> 4, l16 = lane & 15;   // wave32 fragment coords
  const int m0 = (wave >> 2) * 32;               // 0,32
  const int n0 = (wave & 3) * 32;                // 0,32,64,96

  // fallback staging coords
  const int ldrA = tid >> 2, ldcA = (tid & 3) * 4;   // A: 64 rows x 1 float4
  const int ldrW = tid >> 1, ldcW = (tid & 1) * 8;   // W: 128 rows x 2 float4
  const float* Arow = A + (size_t)(bm * 64 + ldrA) * K + ldcA;
  const float* Wrow = W + (size_t)(bn * 128 + ldrW) * K + ldcW;

#ifdef USE_TDM
  const bool issuer = (tid < 32);                // one wave drives the TDM
  const unsigned long long gaA =
      (unsigned long long)(uintptr_t)(A + (size_t)(bm * 64) * K);
  const unsigned long long gaW =
      (unsigned long long)(uintptr_t)(W + (size_t)(bn * 128) * K);
  const i32x8 g1A = tdm_g1(B_ * N_, 64);
  const i32x8 g1W = tdm_g1(NC, 128);
  if (issuer) {                                  // prologue: chunk 0 -> buf 0
    tdm_issue(lds_off(&As[0][0][0]), gaA, g1A);
    tdm_issue(lds_off(&Ws[0][0][0]), gaW, g1W);
    __builtin_amdgcn_s_wait_tensorcnt(0);
  }
  __syncthreads();
#endif

  v8f acc[2][2] = {};
  for (int kk = 0; kk < K; kk += 16) {
    const int cur = (kk >> 4) & 1;
#ifdef USE_TDM
    if (issuer && (kk + 16 < K)) {               // async-fetch next chunk
      tdm_issue(lds_off(&As[cur ^ 1][0][0]),
                gaA + (unsigned long long)(kk + 16) * 4u, g1A);
      tdm_issue(lds_off(&Ws[cur ^ 1][0][0]),
                gaW + (unsigned long long)(kk + 16) * 4u, g1W);
    }
#else
    __syncthreads();
    *(float4*)&As[cur][ldrA][ldcA]     = *(const float4*)(Arow + kk);
    *(float4*)&Ws[cur][ldrW][ldcW]     = *(const float4*)(Wrow + kk);
    *(float4*)&Ws[cur][ldrW][ldcW + 4] = *(const float4*)(Wrow + kk + 4);
    if (kk + 16 < K) {
      __builtin_prefetch(Arow + kk + 16, 0, 1);
      __builtin_prefetch(Wrow + kk + 16, 0, 1);
    }
    __syncthreads();
#endif
    const float(*Ac)[20] = As[cur];
    const float(*Wc)[20] = Ws[cur];
#pragma unroll
    for (int k0 = 0; k0 < 16; k0 += 4) {
      // frags: lanes 0-15 hold K=k0,k0+1 ; lanes 16-31 hold K=k0+2,k0+3
      v2f af[2], bf[2];
      af[0] = *(const v2f*)&Ac[m0 + l16][k0 + 2 * half];
      af[1] = *(const v2f*)&Ac[m0 + 16 + l16][k0 + 2 * half];
      bf[0] = *(const v2f*)&Wc[n0 + l16][k0 + 2 * half];
      bf[1] = *(const v2f*)&Wc[n0 + 16 + l16][k0 + 2 * half];
#pragma unroll
      for (int ms = 0; ms < 2; ++ms)
#pragma unroll
        for (int ns = 0; ns < 2; ++ns)
          acc[ms][ns] = __builtin_amdgcn_wmma_f32_16x16x4_f32(
              false, af[ms], false, bf[ns], (short)0, acc[ms][ns], false, false);
    }
#ifdef USE_TDM
    if (issuer) __builtin_amdgcn_s_wait_tensorcnt(0);
    __syncthreads();                             // publish next buffer
#endif
  }
  // C/D layout: VGPR r, lane l -> M = r + 8*(l>=16), N = l&15
#pragma unroll
  for (int ms = 0; ms < 2; ++ms) {
    const int grow = bm * 64 + m0 + ms * 16 + 8 * half;
#pragma unroll
    for (int ns = 0; ns < 2; ++ns) {
      const int gcol = bn * 128 + n0 + ns * 16 + l16;
      const float bi = bias[gcol];
#pragma unroll
      for (int r = 0; r < 8; ++r)
        Cout[(size_t)(grow + r) * NC + gcol] = acc[ms][ns][r] + bi;
    }
  }
}

// ---------------------------------------------------------------------------
// Per-(b,g): S = (q * N^-1/2)^T k  over tokens (K = 784), then row softmax.
// 288 threads = 9 waves; wave w owns the 32x32 region (w/3, w%3) of 96x96.
// q/k chunks staged TRANSPOSED so fragments are single ds_load_b64s.
// ---------------------------------------------------------------------------
__global__ __launch_bounds__(288) void attn_kernel(const float* __restrict__ qkv,
                                                   float* __restrict__ attn) {
  __shared__ float Qt[96][18];   // [channel][token_in_chunk], padded
  __shared__ float Kt[96][18];
  __shared__ float Ss[96][97];
  const int b = blockIdx.x >> 3, g = blockIdx.x & 7;
  const int tid  = threadIdx.x;
  const int wave = tid >> 5, lane = tid & 31;
  const int half = lane >> 4, l16 = lane & 15;
  const int mb = (wave / 3) * 32;
  const int nb = (wave % 3) * 32;

  const size_t base = (size_t)b * N_ * C3_ + (size_t)g * CH_;
  v8f acc[2][2] = {};

  for (int kk = 0; kk < N_; kk += 16) {  // 49 token chunks
    __syncthreads();
    for (int i = tid; i < 2 * 16 * 96; i += 288) {
      const int mat = i / 1536;                 // 0 = q, 1 = k
      const int rem = i - mat * 1536;
      const int rr = rem / 96, cc = rem % 96;   // rr = token, cc = channel
      const float v = qkv[base + (size_t)(kk + rr) * C3_ + mat * C_ + cc];
      if (mat == 0) Qt[cc][rr] = v; else Kt[cc][rr] = v;
    }
    __syncthreads();
#pragma unroll
    for (int k0 = 0; k0 < 16; k0 += 4) {
      v2f aq[2], bk[2];
#pragma unroll
      for (int s = 0; s < 2; ++s) {
        aq[s] = *(const v2f*)&Qt[mb + s * 16 + l16][k0 + 2 * half];  // A[m=c][k=n]
        bk[s] = *(const v2f*)&Kt[nb + s * 16 + l16][k0 + 2 * half];  // B[k=n][n=d]
      }
#pragma unroll
      for (int ms = 0; ms < 2; ++ms)
#pragma unroll
        for (int ns = 0; ns < 2; ++ns)
          acc[ms][ns] = __builtin_amdgcn_wmma_f32_16x16x4_f32(
              false, aq[ms], false, bk[ns], (short)0, acc[ms][ns], false, false);
    }
  }

  const float scale = 0.035714285714285712f;  // 1/sqrt(784)
#pragma unroll
  for (int ms = 0; ms < 2; ++ms)
#pragma unroll
    for (int ns = 0; ns < 2; ++ns)
#pragma unroll
      for (int r = 0; r < 8; ++r)
        Ss[mb + ms * 16 + r + 8 * half][nb + ns * 16 + l16] =
            acc[ms][ns][r] * scale;
  __syncthreads();

  if (tid < 96) {  // row softmax
    float mx = -3.0e38f;
    for (int d = 0; d < 96; ++d) mx = fmaxf(mx, Ss[tid][d]);
    float sum = 0.f;
    for (int d = 0; d < 96; ++d) {
      const float e = __expf(Ss[tid][d] - mx);
      Ss[tid][d] = e;
      sum += e;
    }
    const float inv = 1.f / sum;
    float* ap = attn + ((size_t)blockIdx.x * 96 + tid) * 96;
    for (int d = 0; d < 96; ++d) ap[d] = Ss[tid][d] * inv;
  }
}

// ---------------------------------------------------------------------------
// y[b, n, g*96+c] = sum_d attn[b,g,c,d] * v[b,g,n,d].  attn cached in LDS.
// 256 threads = 8 waves; each wave sweeps 16-token strips x full 96 channels.
// ---------------------------------------------------------------------------
__global__ __launch_bounds__(256) void apply_kernel(const float* __restrict__ qkv,
                                                    const float* __restrict__ attn,
                                                    float* __restrict__ y) {
  __shared__ float Aw[96][98];
  const int b = blockIdx.x >> 3, g = blockIdx.x & 7;
  const int tid  = threadIdx.x;
  const int wave = tid >> 5, lane = tid & 31;
  const int half = lane >> 4, l16 = lane & 15;

  const float* ap = attn + (size_t)blockIdx.x * 96 * 96;
  for (int i = tid; i < 96 * 96; i += 256) Aw[i / 96][i % 96] = ap[i];
  __syncthreads();

  const size_t vbase = (size_t)b * N_ * C3_ + 2 * C_ + (size_t)g * CH_;
  for (int mt = wave; mt < 49; mt += 8) {
    const int m0 = mt * 16;
    v8f acc[6] = {};
#pragma unroll 4
    for (int k0 = 0; k0 < 96; k0 += 4) {
      // A[m=token][k=d] = v : one aligned float2 global load per lane
      const v2f a =
          *(const v2f*)&qkv[vbase + (size_t)(m0 + l16) * C3_ + k0 + 2 * half];
#pragma unroll
      for (int nt = 0; nt < 6; ++nt) {
        const v2f bb = *(const v2f*)&Aw[nt * 16 + l16][k0 + 2 * half];
        acc[nt] = __builtin_amdgcn_wmma_f32_16x16x4_f32(
            false, a, false, bb, (short)0, acc[nt], false, false);
      }
    }
    const size_t ybase = ((size_t)b * N_ + m0 + 8 * half) * C_ + g * CH_;
#pragma unroll
    for (int nt = 0; nt < 6; ++nt)
#pragma unroll
      for (int r = 0; r < 8; ++r)
        y[ybase + (size_t)r * C_ + nt * 16 + l16] = acc[nt][r];
  }
}

// ---------------------------------------------------------------------------
extern "C" void kernel_launch(void* const* d_in, const int* in_sizes, int n_in,
                              void* d_out, int out_size, void* d_ws, size_t ws_size,
                              hipStream_t stream) {
  const float* x      = (const float*)d_in[0];
  const float* w_qkv  = (const float*)d_in[1];
  const float* b_qkv  = (const float*)d_in[2];
  const float* w_proj = (const float*)d_in[3];
  const float* b_proj = (const float*)d_in[4];
  const void*  sz_in  = d_in[5];
  float* out = (float*)d_out;

  // workspace: qkv [B,N,3C] | attn [B,G,96,96] | y [B,N,C]   (~318 MB f32)
  float* qkv  = (float*)d_ws;
  float* attn = qkv + (size_t)B_ * N_ * C3_;
  float* y    = attn + (size_t)B_ * G_ * CH_ * CH_;

  gemm_bias_kernel<C3_><<<dim3(C3_ / 128, (B_ * N_) / 64), 256, 0, stream>>>(
      x, w_qkv, b_qkv, qkv);
  attn_kernel<<<B_ * G_, 288, 0, stream>>>(qkv, attn);
  apply_kernel<<<B_ * G_, 256, 0, stream>>>(qkv, attn, y);
  gemm_bias_kernel<C_><<<dim3(C_ / 128, (B_ * N_) / 64), 256, 0, stream>>>(
      y, w_proj, b_proj, out);

  // tuple tail: pass `size` (2 x int32) through bit-exactly
  hipMemcpyAsync((char*)d_out + (size_t)B_ * N_ * C_ * sizeof(float), sz_in,
                 2 * sizeof(int), hipMemcpyDeviceToDevice, stream);
}